// RandomTopKModifier_43396349559002
// MI455X (gfx1250) — compile-verified
//
#include <hip/hip_runtime.h>
#include <math.h>

// ---------------------------------------------------------------------------
// RandomTopKModifier forward for MI455X (gfx1250, wave32).
//
// Pipeline (all streaming, HBM-bound; ~640MB total traffic ~ 27us @ 23.3TB/s):
//   1) 3-level radix select (11/11/10 bits) on orderable |x| bits -> thrA[row]
//   2) one pass: exact JAX threefry2x32-20 (key = (0,42)) -> gumbel -> score,
//      packed as orderable u32 into d_out (scratch; fully overwritten later)
//   3) 3-level radix select on score keys -> thrS[row]
//   4) final pass: out = (score_key >= thrS) ? x : 0
//
// CDNA5 specifics:
//   - async global->LDS staging (ASYNCcnt) with double buffering via
//     global_load_async_to_lds_b128 + s_wait_asynccnt (builtins confirmed
//     present by round-1 diagnostics; params are v4i AS1* / v4i AS3*)
//   - global_prefetch_b8 via __builtin_prefetch in histogram passes
//   - wave32-aware block layout (256 threads = 8 waves, wave-private staging)
// ---------------------------------------------------------------------------

typedef unsigned int u32;

#ifndef __has_builtin
#define __has_builtin(x) 0
#endif

#if __has_builtin(__builtin_amdgcn_global_load_async_to_lds_b128) && \
    __has_builtin(__builtin_amdgcn_s_wait_asynccnt)
#define RTK_ASYNC_LDS 1
#else
#define RTK_ASYNC_LDS 0
#endif

#if RTK_ASYNC_LDS
typedef int rtk_v4i __attribute__((ext_vector_type(4)));
typedef __attribute__((address_space(1))) rtk_v4i rtk_v4i_glob;  // global
typedef __attribute__((address_space(3))) rtk_v4i rtk_v4i_lds;   // LDS
__device__ __forceinline__ void rtk_async_b128(void* lds, const void* g) {
  __builtin_amdgcn_global_load_async_to_lds_b128(
      (rtk_v4i_glob*)(g), (rtk_v4i_lds*)(lds), 0, 0);
}
#endif

// -------------------------- RNG: JAX threefry ------------------------------

__device__ __forceinline__ u32 rtk_rotl(u32 x, int r) {
  return (x << r) | (x >> (32 - r));
}

__device__ __forceinline__ void rtk_tf4(u32& x0, u32& x1, int r0, int r1, int r2, int r3) {
  x0 += x1; x1 = rtk_rotl(x1, r0); x1 ^= x0;
  x0 += x1; x1 = rtk_rotl(x1, r1); x1 ^= x0;
  x0 += x1; x1 = rtk_rotl(x1, r2); x1 ^= x0;
  x0 += x1; x1 = rtk_rotl(x1, r3); x1 ^= x0;
}

// JAX random_bits for a flat array of 2*half elements with key (0, 42):
// element i pairs counter (i mod half, i mod half + half); takes x0 if
// i < half else x1. threefry-2x32, 20 rounds.
__device__ __forceinline__ u32 rtk_threefry_bits(u32 i, u32 half) {
  u32 j  = (i < half) ? i : (i - half);
  u32 x0 = j, x1 = j + half;
  const u32 k0 = 0u, k1 = 42u, k2 = 0x1BD11BDAu ^ k0 ^ k1;
  x0 += k0; x1 += k1;
  rtk_tf4(x0, x1, 13, 15, 26, 6);  x0 += k1; x1 += k2 + 1u;
  rtk_tf4(x0, x1, 17, 29, 16, 24); x0 += k2; x1 += k0 + 2u;
  rtk_tf4(x0, x1, 13, 15, 26, 6);  x0 += k0; x1 += k1 + 3u;
  rtk_tf4(x0, x1, 17, 29, 16, 24); x0 += k1; x1 += k2 + 4u;
  rtk_tf4(x0, x1, 13, 15, 26, 6);  x0 += k2; x1 += k0 + 5u;
  return (i < half) ? x0 : x1;
}

// Monotone float -> u32 transform (top-k of float == top-k of key).
__device__ __forceinline__ u32 rtk_flip(u32 u) {
  return (u & 0x80000000u) ? ~u : (u | 0x80000000u);
}

// score key: gumbel(rng(idx)) + log(prob(|x| topk?)) packed orderable.
__device__ __forceinline__ u32 rtk_score_key(float xv, u32 flat_idx, u32 thrA,
                                             float logPhi, float logPlo, u32 half) {
  u32 ab  = __float_as_uint(fabsf(xv)) | 0x80000000u;  // flip() for >=0 floats
  u32 rb  = rtk_threefry_bits(flat_idx, half);
  float f = __uint_as_float((rb >> 9) | 0x3f800000u) - 1.0f;   // [0,1)
  f       = fmaxf(f, 1.1754944e-38f);
  float g = -__logf(-__logf(f));
  float s = g + ((ab >= thrA) ? logPhi : logPlo);
  return rtk_flip(__float_as_uint(s));
}

// -------------------------- sizes ------------------------------------------

#define RTK_B     32
#define RTK_BINS  2048
#define RTK_ITER  8      // float4 iterations per block: 8*256*4 = 8192 elems

// -------------------------------- kernels ----------------------------------

__global__ __launch_bounds__(256) void rtk_zero_hist(u32* hist) {
  hist[(size_t)blockIdx.x * blockDim.x + threadIdx.x] = 0u;
}

__global__ void rtk_init_state(u32* prefix, u32* kremain, u32 k) {
  prefix[threadIdx.x]  = 0u;
  kremain[threadIdx.x] = k;
}

// Per-row radix histogram pass. mode 0: keys computed from |x|; mode 1: keys
// streamed from precomputed score-key buffer.
__global__ __launch_bounds__(256) void rtk_hist(const float* __restrict__ x,
                                                const u32* __restrict__ keys,
                                                u32* __restrict__ hist,
                                                const u32* __restrict__ prefix,
                                                int n, int shift, int width,
                                                int mode, int level) {
  __shared__ u32 lh[RTK_BINS];
  for (int i = threadIdx.x; i < RTK_BINS; i += blockDim.x) lh[i] = 0u;
  __syncthreads();

  const int row  = blockIdx.y;
  const u32 pfx  = prefix[row];
  const u32 mask = (1u << width) - 1u;
  const size_t base4 = (size_t)row * (size_t)(n >> 2) +
                       (size_t)blockIdx.x * (256 * RTK_ITER) + threadIdx.x;

  if (mode == 0) {
    const float4* g = (const float4*)x + base4;
#pragma unroll
    for (int it = 0; it < RTK_ITER; ++it) {
      if (it + 1 < RTK_ITER) __builtin_prefetch(g + (it + 1) * 256, 0, 1);
      float4 v = g[it * 256];
      u32 u[4] = { __float_as_uint(fabsf(v.x)) | 0x80000000u,
                   __float_as_uint(fabsf(v.y)) | 0x80000000u,
                   __float_as_uint(fabsf(v.z)) | 0x80000000u,
                   __float_as_uint(fabsf(v.w)) | 0x80000000u };
#pragma unroll
      for (int c = 0; c < 4; ++c) {
        bool ok = (level == 0) || ((u[c] >> (shift + width)) == pfx);
        if (ok) atomicAdd(&lh[(u[c] >> shift) & mask], 1u);
      }
    }
  } else {
    const uint4* g = (const uint4*)keys + base4;
#pragma unroll
    for (int it = 0; it < RTK_ITER; ++it) {
      if (it + 1 < RTK_ITER) __builtin_prefetch(g + (it + 1) * 256, 0, 1);
      uint4 v = g[it * 256];
      u32 u[4] = { v.x, v.y, v.z, v.w };
#pragma unroll
      for (int c = 0; c < 4; ++c) {
        bool ok = (level == 0) || ((u[c] >> (shift + width)) == pfx);
        if (ok) atomicAdd(&lh[(u[c] >> shift) & mask], 1u);
      }
    }
  }

  __syncthreads();
  for (int i = threadIdx.x; i < (1 << width); i += blockDim.x)
    if (lh[i]) atomicAdd(&hist[(size_t)row * RTK_BINS + i], lh[i]);
}

// One block per row: walk histogram from the top bin, find the bin holding the
// k-th largest key, update (prefix, kremain), zero the histogram for reuse.
__global__ __launch_bounds__(256) void rtk_scan(u32* hist, u32* prefix, u32* kremain,
                                                u32* thrOut, int width, int last,
                                                u32 kInit) {
  const int row = blockIdx.x;
  u32* h = hist + (size_t)row * RTK_BINS;
  if (threadIdx.x == 0) {
    u32 kr = kremain[row];
    int nb = 1 << width;
    u32 c = 0; int sel = 0; u32 above = 0; bool found = false;
    for (int b = nb - 1; b >= 0; --b) {
      u32 nc = c + h[b];
      if (nc >= kr) { sel = b; above = c; found = true; break; }
      c = nc;
    }
    if (!found) { sel = 0; above = c; }
    u32 pfx = (prefix[row] << width) | (u32)sel;
    if (last) {
      thrOut[row]  = pfx;       // full 32-bit threshold (11+11+10 bits)
      prefix[row]  = 0u;        // reset state for the next selection phase
      kremain[row] = kInit;
    } else {
      prefix[row]  = pfx;
      kremain[row] = kr - above;
    }
  }
  __syncthreads();
  for (int i = threadIdx.x; i < RTK_BINS; i += blockDim.x) h[i] = 0u;
}

// Compute score keys once (RNG is the expensive part) -> sbits (== d_out).
// Streams x through LDS via the gfx1250 async path (ASYNCcnt double buffer).
__global__ __launch_bounds__(256) void rtk_score_init(const float* __restrict__ x,
                                                      u32* __restrict__ sbits,
                                                      const u32* __restrict__ thrA,
                                                      int n, float logPhi, float logPlo,
                                                      u32 half) {
  const int row  = blockIdx.y;
  const int lane = threadIdx.x & 31;
  const int wave = threadIdx.x >> 5;
  const u32 thr  = thrA[row];
  const size_t base4 = (size_t)row * (size_t)(n >> 2) +
                       (size_t)blockIdx.x * (256 * RTK_ITER) + threadIdx.x;
  const float4* gx = (const float4*)x + base4;
  uint4* gs        = (uint4*)sbits + base4;

#if RTK_ASYNC_LDS
  __shared__ float4 stage[8][2][32];   // wave-private double buffers
  rtk_async_b128(&stage[wave][0][lane], gx);
#endif
#pragma unroll
  for (int it = 0; it < RTK_ITER; ++it) {
    float4 v;
#if RTK_ASYNC_LDS
    if (it + 1 < RTK_ITER) {
      rtk_async_b128(&stage[wave][(it + 1) & 1][lane], gx + (it + 1) * 256);
      __builtin_amdgcn_s_wait_asynccnt(1);
    } else {
      __builtin_amdgcn_s_wait_asynccnt(0);
    }
    asm volatile("" ::: "memory");
    v = stage[wave][it & 1][lane];
#else
    v = gx[it * 256];
#endif
    u32 q    = (u32)blockIdx.x * (256u * RTK_ITER) + (u32)it * 256u + (u32)threadIdx.x;
    u32 fidx = (u32)row * (u32)n + q * 4u;
    uint4 sb;
    sb.x = rtk_score_key(v.x, fidx + 0u, thr, logPhi, logPlo, half);
    sb.y = rtk_score_key(v.y, fidx + 1u, thr, logPhi, logPlo, half);
    sb.z = rtk_score_key(v.z, fidx + 2u, thr, logPhi, logPlo, half);
    sb.w = rtk_score_key(v.w, fidx + 3u, thr, logPhi, logPlo, half);
    gs[it * 256] = sb;
  }
}

// Final mask pass: out = (score_key >= thrS[row]) ? x : 0.
// Reads keys from d_out, overwrites d_out in place (same thread owns both).
__global__ __launch_bounds__(256) void rtk_final(const float* __restrict__ x,
                                                 float* __restrict__ out,
                                                 const u32* __restrict__ thrS,
                                                 int n) {
  const int row  = blockIdx.y;
  const int lane = threadIdx.x & 31;
  const int wave = threadIdx.x >> 5;
  const u32 thr  = thrS[row];
  const size_t base4 = (size_t)row * (size_t)(n >> 2) +
                       (size_t)blockIdx.x * (256 * RTK_ITER) + threadIdx.x;
  const float4* gx = (const float4*)x + base4;
  const uint4* gs  = (const uint4*)out + base4;
  float4* go       = (float4*)out + base4;

#if RTK_ASYNC_LDS
  __shared__ float4 stage[8][2][32];
  rtk_async_b128(&stage[wave][0][lane], gx);
#endif
#pragma unroll
  for (int it = 0; it < RTK_ITER; ++it) {
    uint4 s = gs[it * 256];
    float4 v;
#if RTK_ASYNC_LDS
    if (it + 1 < RTK_ITER) {
      rtk_async_b128(&stage[wave][(it + 1) & 1][lane], gx + (it + 1) * 256);
      __builtin_amdgcn_s_wait_asynccnt(1);
    } else {
      __builtin_amdgcn_s_wait_asynccnt(0);
    }
    asm volatile("" ::: "memory");
    v = stage[wave][it & 1][lane];
#else
    v = gx[it * 256];
#endif
    float4 o;
    o.x = (s.x >= thr) ? v.x : 0.0f;
    o.y = (s.y >= thr) ? v.y : 0.0f;
    o.z = (s.z >= thr) ? v.z : 0.0f;
    o.w = (s.w >= thr) ? v.w : 0.0f;
    go[it * 256] = o;
  }
}

// ------------------------------ host launch --------------------------------

extern "C" void kernel_launch(void* const* d_in, const int* in_sizes, int n_in,
                              void* d_out, int out_size, void* d_ws, size_t ws_size,
                              hipStream_t stream) {
  const float* x  = (const float*)d_in[0];
  const int total = in_sizes[0];          // 32 * 524288
  const int B     = RTK_B;
  const int n     = total / B;            // 524288 (multiple of 8192)
  const u32 k     = (u32)(0.1 * (double)n + 0.5);   // 52429

  u32* hist    = (u32*)d_ws;              // B * 2048 u32 = 256 KB
  u32* prefix  = hist + (size_t)B * RTK_BINS;
  u32* kremain = prefix + B;
  u32* thrA    = kremain + B;
  u32* thrS    = thrA + B;
  u32* sbits   = (u32*)d_out;             // score keys staged in d_out

  const float logPhi = logf((1.0f - 0.1f) / (float)k);
  const float logPlo = logf(0.1f / (float)(n - (int)k));
  const u32 half     = (u32)(total / 2);

  dim3 sgrid((unsigned)(n / (256 * RTK_ITER * 4)), (unsigned)B);  // 64 x 32
  const int shifts[3] = {21, 10, 0};
  const int widths[3] = {11, 11, 10};

  rtk_zero_hist<<<(B * RTK_BINS) / 256, 256, 0, stream>>>(hist);
  rtk_init_state<<<1, B, 0, stream>>>(prefix, kremain, k);

  // phase 1: threshold on |x|
  for (int L = 0; L < 3; ++L) {
    rtk_hist<<<sgrid, 256, 0, stream>>>(x, sbits, hist, prefix, n,
                                        shifts[L], widths[L], /*mode=*/0, L);
    rtk_scan<<<B, 256, 0, stream>>>(hist, prefix, kremain, thrA,
                                    widths[L], L == 2, k);
  }

  // phase 2: score keys (RNG once), then threshold on scores
  rtk_score_init<<<sgrid, 256, 0, stream>>>(x, sbits, thrA, n, logPhi, logPlo, half);
  for (int L = 0; L < 3; ++L) {
    rtk_hist<<<sgrid, 256, 0, stream>>>(x, sbits, hist, prefix, n,
                                        shifts[L], widths[L], /*mode=*/1, L);
    rtk_scan<<<B, 256, 0, stream>>>(hist, prefix, kremain, thrS,
                                    widths[L], L == 2, k);
  }

  // phase 3: apply mask
  rtk_final<<<sgrid, 256, 0, stream>>>(x, (float*)d_out, thrS, n);
}